// RNN_classifier_20117626815076
// MI455X (gfx1250) — compile-verified
//
#include <hip/hip_runtime.h>

#define LL    64
#define BB    2048
#define DD    512
#define VPT   4                 // d's per thread
#define TPB   (DD / VPT)        // 128 threads = 4 waves (wave32)
#define NWAVE (TPB / 32)        // 4

typedef __attribute__((ext_vector_type(2))) float v2f;
typedef __attribute__((ext_vector_type(8))) float v8f;

// ---- fast transcendental helpers (native CDNA5 TRANS ops) ----
__device__ __forceinline__ float fast_rcp(float x) {
#if __has_builtin(__builtin_amdgcn_rcpf)
    return __builtin_amdgcn_rcpf(x);        // v_rcp_f32
#else
    return 1.0f / x;
#endif
}

__device__ __forceinline__ float fast_tanh(float x) {
#if __has_builtin(__builtin_amdgcn_tanhf)
    return __builtin_amdgcn_tanhf(x);       // v_tanh_f32 (gfx1250)
#elif __has_builtin(__builtin_amdgcn_tanh_f32)
    return __builtin_amdgcn_tanh_f32(x);
#else
    return tanhf(x);
#endif
}

__device__ __forceinline__ float sigm(float z) {
    return fast_rcp(1.0f + __expf(-z));     // v_mul + v_exp + v_add + v_rcp
}

__global__ __launch_bounds__(TPB)
void rnn_fused_kernel(const float* __restrict__ x,
                      const float* __restrict__ wx,
                      const float* __restrict__ wg,
                      const float* __restrict__ wh,
                      const float* __restrict__ wu,
                      const float* __restrict__ msk,
                      const float* __restrict__ fcw,
                      const float* __restrict__ fcb,
                      float* __restrict__ out)
{
    const int b    = blockIdx.x;
    const int tid  = threadIdx.x;
    const int lane = tid & 31;
    const int wave = tid >> 5;

    __shared__ float partial[NWAVE][LL];   // wave-private rows -> deterministic

    const float4* xv  = (const float4*)x;
    const float4* wxv = (const float4*)wx;
    const float4* wgv = (const float4*)wg;
    const float4* whv = (const float4*)wh;
    const float4* wuv = (const float4*)wu;
    const float4* mv  = (const float4*)msk;

    const int d4      = DD / 4;
    const int xstride = BB * d4;           // float4 stride per time step
    const int xbase   = b * d4 + tid;

    float4 fw4 = ((const float4*)fcw)[tid];
    const float* fw = (const float*)&fw4;

    float h[VPT], h0v[VPT], xb[VPT];

    // ---- t = 0 bootstrap: x_bar[0] = x[0]; h0 = tanh(x0*wx0); x_bar[1] = h0*wg0 + x0
    float4 x4 = xv[xbase];
    float4 a4 = wxv[tid];
    float4 g4 = wgv[tid];
    const float* xp = (const float*)&x4;
    const float* ap = (const float*)&a4;
    const float* gp = (const float*)&g4;

    float c = 0.f;
    #pragma unroll
    for (int i = 0; i < VPT; ++i) {
        float hx = fast_tanh(xp[i] * ap[i]);
        h[i] = hx; h0v[i] = hx;
        xb[i] = hx * gp[i] + xp[i];
        c += xp[i] * fw[i];                 // x_bar[0] dot contribution
    }
    #pragma unroll
    for (int off = 16; off > 0; off >>= 1) c += __shfl_down(c, off, 32);
    if (lane == 0) partial[wave][0] = c;

    // ---- t = 1 .. 62: emit x_bar[t], then advance with x[t], w[t]
    for (int t = 1; t < LL - 1; ++t) {
        c = 0.f;
        #pragma unroll
        for (int i = 0; i < VPT; ++i) c += xb[i] * fw[i];
        #pragma unroll
        for (int off = 16; off > 0; off >>= 1) c += __shfl_down(c, off, 32);
        if (lane == 0) partial[wave][t] = c;

        float4 xt4 = xv[t * xstride + xbase];
        __builtin_prefetch(&xv[(t + 1) * xstride + xbase], 0, 3); // global_prefetch_b8
        float4 m4  = mv [t * d4 + tid];
        float4 wx4 = wxv[t * d4 + tid];
        float4 wh4 = whv[t * d4 + tid];
        float4 wu4 = wuv[t * d4 + tid];
        float4 wg4 = wgv[t * d4 + tid];
        const float* xtp = (const float*)&xt4;
        const float* mp  = (const float*)&m4;
        const float* wxp = (const float*)&wx4;
        const float* whp = (const float*)&wh4;
        const float* wup = (const float*)&wu4;
        const float* wgp = (const float*)&wg4;

        #pragma unroll
        for (int i = 0; i < VPT; ++i) {
            float xtr = xb[i] + mp[i] * (xtp[i] - xb[i]);         // lerp
            float u   = fast_tanh(xtr * wxp[i]);
            float f   = sigm(h[i] * whp[i] + u * wup[i]) * mp[i];
            float hn  = f * h[i] + fminf(1.0f - f, u);
            h[i]      = fmaxf(hn, h0v[i]);                        // h0 + relu(h-h0)
            xb[i]     = h[i] * wgp[i] + xtr;
        }
    }

    // ---- emit x_bar[63]
    c = 0.f;
    #pragma unroll
    for (int i = 0; i < VPT; ++i) c += xb[i] * fw[i];
    #pragma unroll
    for (int off = 16; off > 0; off >>= 1) c += __shfl_down(c, off, 32);
    if (lane == 0) partial[wave][LL - 1] = c;

    __syncthreads();

    // ---- cross-wave combine via V_WMMA_F32_16X16X4_F32:
    // A(16x4): A[m][k] = partial[k][16*wave + m]; B = ones  =>  D[m][*] = sum_k partial[k][...]
    const int grp = lane >> 4;          // 0: holds K=0,1   1: holds K=2,3 (ISA A layout)
    const int row = lane & 15;          // M index
    const int l16 = 16 * wave;
    v2f A;  A[0] = partial[2 * grp + 0][l16 + row];
            A[1] = partial[2 * grp + 1][l16 + row];
    v2f Bm; Bm[0] = 1.0f; Bm[1] = 1.0f;
    v8f Cm = {};
    v8f Dm = __builtin_amdgcn_wmma_f32_16x16x4_f32(
        false, A, false, Bm, (short)0, Cm, false, false);

    const float bias = fcb[0];
    if (row == 0) {                     // lanes 0 (M=0..7) and 16 (M=8..15)
        const int mbase = grp * 8;      // D layout: VGPR r -> M=r (lanes 0-15), M=8+r (16-31)
        #pragma unroll
        for (int r = 0; r < 8; ++r) {
            const int l = l16 + mbase + r;
            out[l * BB + b] = sigm(Dm[r] + bias);
        }
    }
}

extern "C" void kernel_launch(void* const* d_in, const int* in_sizes, int n_in,
                              void* d_out, int out_size, void* d_ws, size_t ws_size,
                              hipStream_t stream) {
    const float* x   = (const float*)d_in[0];
    const float* wx  = (const float*)d_in[1];
    const float* wg  = (const float*)d_in[2];
    const float* wh  = (const float*)d_in[3];
    const float* wu  = (const float*)d_in[4];
    const float* msk = (const float*)d_in[5];
    const float* fcw = (const float*)d_in[6];
    const float* fcb = (const float*)d_in[7];
    float* out = (float*)d_out;
    rnn_fused_kernel<<<dim3(BB), dim3(TPB), 0, stream>>>(
        x, wx, wg, wh, wu, msk, fcw, fcb, out);
}